// attention_node_to_edge_12034498363709
// MI455X (gfx1250) — compile-verified
//
#include <hip/hip_runtime.h>

typedef float v2f __attribute__((ext_vector_type(2)));
typedef float v4f __attribute__((ext_vector_type(4)));
typedef float v8f __attribute__((ext_vector_type(8)));

#define BB   8
#define NN   256
#define DD   256
#define DHH  64
#define MTILE 32          // rows of e processed per LDS stage
#define ES_STRIDE 260     // 256 + 4 pad -> conflict-free b64 A-fragment loads
#define VS_STRIDE 65      // 64 + 1 pad  -> conflict-free V stores

// ---------------------------------------------------------------------------
// Kernel 1: Q = (x @ Wq + bq) * node_mask   (tiny: 33 MFLOP total)
// Written into d_out[0 : B*N*DH] which the fused kernel reads back and then
// overwrites with x_out.
// ---------------------------------------------------------------------------
__global__ __launch_bounds__(64)
void qproj_kernel(const float* __restrict__ x, const float* __restrict__ node_mask,
                  const float* __restrict__ Wq, const float* __restrict__ bq,
                  float* __restrict__ Qout) {
  __shared__ float xs[DD];
  const int bn = blockIdx.x;     // b*N + n
  const int h  = threadIdx.x;    // 0..63
  #pragma unroll
  for (int i = 0; i < DD / 64; ++i)
    xs[h + i * 64] = x[(size_t)bn * DD + h + i * 64];
  __syncthreads();
  float acc = bq[h];
  #pragma unroll 4
  for (int d = 0; d < DD; ++d)
    acc += xs[d] * Wq[d * DHH + h];
  Qout[(size_t)bn * DHH + h] = acc * node_mask[bn];
}

// ---------------------------------------------------------------------------
// Kernel 2: per-(b,n) fused  K/V GEMM (f32 WMMA) + scores + softmax + Att@V,
// with write-through copy of e into d_out. K and V never touch HBM.
// Waves 0-3: K matrix (h-tiles 0..3). Waves 4-7: V matrix (h-tiles 0..3).
// Each wave keeps its 256x16 slice of W as 64 v2f B-fragments in registers.
// Four independent WMMA accumulator chains hide the WMMA RAW latency.
// ---------------------------------------------------------------------------
__global__ __launch_bounds__(256)
void fused_edge_attn_kernel(const float* __restrict__ e,
                            const float* __restrict__ node_mask,
                            const float* __restrict__ Wk, const float* __restrict__ bk,
                            const float* __restrict__ Wv, const float* __restrict__ bv,
                            float* __restrict__ qx,      // in: Q row, out: x_out row
                            float* __restrict__ e_out) {
  __shared__ float e_s[MTILE][ES_STRIDE];   // 33.3 KB
  __shared__ float v_s[NN][VS_STRIDE];      // 66.6 KB (full V row, bias included)
  __shared__ float sc[NN];                  // raw scores  Q.K + Q.bk
  __shared__ float red[256];
  __shared__ float q_s[DHH];

  const int bn   = blockIdx.x;             // b*N + n
  const int b    = bn >> 8;                // N = 256
  const int tid  = threadIdx.x;
  const int wave = tid >> 5;
  const int lane = tid & 31;
  const int lc   = lane & 15;
  const int half = lane >> 4;
  const int mat  = wave >> 2;              // 0 = K, 1 = V (wave-uniform)
  const int h0   = (wave & 3) * 16;
  const int hcol = h0 + lc;

  const float* __restrict__ Wm = mat ? Wv : Wk;
  const float* __restrict__ bm = mat ? bv : bk;

  // --- preload this wave's B fragments (4x16 tiles of W, k-major) ---------
  // B layout (16x16x4): VGPR r, lanes 0-15 -> K=r, lanes 16-31 -> K=r+2.
  v2f bfrag[64];
  #pragma unroll
  for (int kk = 0; kk < 64; ++kk) {
    const int krow = 4 * kk + 2 * half;
    v2f t;
    t.x = Wm[(size_t)krow * DHH + hcol];
    t.y = Wm[(size_t)(krow + 1) * DHH + hcol];
    bfrag[kk] = t;
  }
  const float bias = bm[hcol];             // folded into accumulator init

  sc[tid] = 0.0f;
  if (tid < DHH) q_s[tid] = qx[(size_t)bn * DHH + tid];
  __syncthreads();

  const float* __restrict__ e_row = e + (size_t)bn * NN * DD;
  float* __restrict__ eo_row      = e_out + (size_t)bn * NN * DD;

  for (int mt = 0; mt < NN / MTILE; ++mt) {
    // --- stage e tile (32 x 256) into LDS; write-through copy to d_out ---
    const v4f* __restrict__ src = (const v4f*)(e_row + (size_t)mt * MTILE * DD);
    v4f* __restrict__ dst       = (v4f*)(eo_row + (size_t)mt * MTILE * DD);
    #pragma unroll
    for (int i = 0; i < (MTILE * DD / 4) / 256; ++i) {   // 8 x float4 / thread
      const int idx4 = tid + i * 256;
      const v4f v = __builtin_nontemporal_load(&src[idx4]);
      __builtin_nontemporal_store(v, &dst[idx4]);
      const int row = idx4 >> 6;           // / (DD/4)
      const int col = (idx4 & 63) << 2;
      *(v4f*)&e_s[row][col] = v;
    }
    __syncthreads();

    // --- K/V GEMM: two 16-row subtiles x two k-partials = 4 WMMA chains --
    v8f acc0a, acc1a, acc0b, acc1b;
    #pragma unroll
    for (int r = 0; r < 8; ++r) {
      acc0a[r] = bias; acc1a[r] = bias;    // bias folded once (chain a)
      acc0b[r] = 0.0f; acc1b[r] = 0.0f;
    }

    #pragma unroll
    for (int kk = 0; kk < 64; kk += 2) {
      const int kc0 = 4 * kk + 2 * half;       // A: lanes 0-15 -> K0,K1; 16-31 -> K2,K3
      const int kc1 = kc0 + 4;
      const v2f a0a = *(const v2f*)&e_s[lc][kc0];
      const v2f a1a = *(const v2f*)&e_s[16 + lc][kc0];
      const v2f a0b = *(const v2f*)&e_s[lc][kc1];
      const v2f a1b = *(const v2f*)&e_s[16 + lc][kc1];
      acc0a = __builtin_amdgcn_wmma_f32_16x16x4_f32(false, a0a, false, bfrag[kk],
                                                    (short)0, acc0a, false, false);
      acc1a = __builtin_amdgcn_wmma_f32_16x16x4_f32(false, a1a, false, bfrag[kk],
                                                    (short)0, acc1a, false, false);
      acc0b = __builtin_amdgcn_wmma_f32_16x16x4_f32(false, a0b, false, bfrag[kk + 1],
                                                    (short)0, acc0b, false, false);
      acc1b = __builtin_amdgcn_wmma_f32_16x16x4_f32(false, a1b, false, bfrag[kk + 1],
                                                    (short)0, acc1b, false, false);
    }

    v8f acc0, acc1;
    #pragma unroll
    for (int r = 0; r < 8; ++r) {
      acc0[r] = acc0a[r] + acc0b[r];
      acc1[r] = acc1a[r] + acc1b[r];
    }

    if (mat == 0) {
      // scores: sc[m] += sum_h q[h] * K[m][h]   (C/D: VGPR r -> row r / r+8)
      #pragma unroll
      for (int r = 0; r < 8; ++r) {
        float p0 = q_s[hcol] * acc0[r];
        float p1 = q_s[hcol] * acc1[r];
        #pragma unroll
        for (int off = 1; off < 16; off <<= 1) {
          p0 += __shfl_xor(p0, off, 32);
          p1 += __shfl_xor(p1, off, 32);
        }
        if (lc == 0) {
          const int m0 = mt * MTILE + r + 8 * half;
          atomicAdd(&sc[m0], p0);
          atomicAdd(&sc[m0 + 16], p1);
        }
      }
    } else {
      #pragma unroll
      for (int r = 0; r < 8; ++r) {
        v_s[mt * MTILE + r + 8 * half][hcol]      = acc0[r];
        v_s[mt * MTILE + 16 + r + 8 * half][hcol] = acc1[r];
      }
    }
    __syncthreads();   // e_s reused next iter; sc/v_s complete at loop end
  }

  // --- masked, numerically-stable softmax over m (256 threads = 256 m) ----
  const float maskn = node_mask[bn];
  const float maskm = node_mask[b * NN + tid];
  const float s = (maskn * maskm > 0.0f) ? sc[tid] * 0.125f : -1.0e9f; // /sqrt(64)
  red[tid] = s;
  __syncthreads();
  #pragma unroll
  for (int st = 128; st > 0; st >>= 1) {
    if (tid < st) red[tid] = fmaxf(red[tid], red[tid + st]);
    __syncthreads();
  }
  const float smax = red[0];
  __syncthreads();
  const float p = __expf(s - smax);
  red[tid] = p;
  __syncthreads();
  #pragma unroll
  for (int st = 128; st > 0; st >>= 1) {
    if (tid < st) red[tid] += red[tid + st];
    __syncthreads();
  }
  const float denom = red[0];
  __syncthreads();
  sc[tid] = (p / denom) * maskm;   // fold V's m-mask into attention weight
  __syncthreads();

  // --- x_out[h] = maskn * sum_m attw[m] * V[m][h] -------------------------
  const int h  = tid & 63;
  const int c4 = tid >> 6;         // 4 partial chunks of 64 m each
  float part = 0.0f;
  #pragma unroll 8
  for (int j = 0; j < 64; ++j) {
    const int m = c4 * 64 + j;
    part += sc[m] * v_s[m][h];
  }
  red[tid] = part;
  __syncthreads();
  if (tid < 64) {
    const float xo = red[tid] + red[tid + 64] + red[tid + 128] + red[tid + 192];
    qx[(size_t)bn * DHH + tid] = maskn * xo;   // overwrites Q with x_out
  }
}

// ---------------------------------------------------------------------------
extern "C" void kernel_launch(void* const* d_in, const int* in_sizes, int n_in,
                              void* d_out, int out_size, void* d_ws, size_t ws_size,
                              hipStream_t stream) {
  (void)in_sizes; (void)n_in; (void)d_ws; (void)ws_size; (void)out_size;
  const float* x  = (const float*)d_in[0];
  const float* e  = (const float*)d_in[1];
  const float* nm = (const float*)d_in[2];
  const float* Wq = (const float*)d_in[3];
  const float* bq = (const float*)d_in[4];
  const float* Wk = (const float*)d_in[5];
  const float* bk = (const float*)d_in[6];
  const float* Wv = (const float*)d_in[7];
  const float* bv = (const float*)d_in[8];

  float* out  = (float*)d_out;
  float* xout = out;                                   // B*N*DH floats
  float* eout = out + (size_t)BB * NN * DHH;           // copy of e

  qproj_kernel<<<BB * NN, 64, 0, stream>>>(x, nm, Wq, bq, xout);
  fused_edge_attn_kernel<<<BB * NN, 256, 0, stream>>>(e, nm, Wk, bk, Wv, bv,
                                                      xout, eout);
}